// AllMixup_57251914056261
// MI455X (gfx1250) — compile-verified
//
#include <hip/hip_runtime.h>
#include <hip/hip_bf16.h>

// Masked one-hot fill:  out[row, c] = (labels[row] == c && labels[row] >= 0) ? 1 : 0
// B=256, N=128, C=607  ->  rows = 32768, out_size = 19,890,176 f32 (79.5 MB).
// Pure store-bandwidth problem (zero FLOPs): bulk zero-fill uses the gfx1250
// async LDS->global store path (GLOBAL_STORE_ASYNC_FROM_LDS_B128, ASYNCcnt),
// then a tiny scatter kernel writes the ones. Roofline: 79.5 MB / 23.3 TB/s
// ~= 3.4 us; output even fits in the 192 MB L2.

typedef float float4v __attribute__((ext_vector_type(4)));
typedef int   v4i     __attribute__((ext_vector_type(4)));

#if defined(__AMDGCN__)
typedef __attribute__((address_space(1))) v4i* gv4i_ptr;   // global v4i*
typedef __attribute__((address_space(3))) v4i* lv4i_ptr;   // LDS v4i*
#endif

__global__ void onehot_zero_fill(float* __restrict__ out,
                                 long long n_vec4, int rem) {
    __shared__ float lds_zero[256 * 4];   // 16 B per thread, all zeros
    const int tid = threadIdx.x;

    // Zero this thread's 16-byte LDS slot (source of the async stores).
    float4v z = {0.f, 0.f, 0.f, 0.f};
    *(float4v*)(&lds_zero[tid * 4]) = z;
    __syncthreads();

    const long long stride = (long long)gridDim.x * blockDim.x;
    long long i = (long long)blockIdx.x * blockDim.x + tid;
    float4v* __restrict__ out4 = (float4v*)out;

#if defined(__AMDGCN__) && __has_builtin(__builtin_amdgcn_global_store_async_from_lds_b128)
    lv4i_ptr lsrc = (lv4i_ptr)(&lds_zero[tid * 4]);
    for (; i < n_vec4; i += stride) {
        // 16 B per lane, 512 B per wave32 instruction; tracked by ASYNCcnt,
        // no VGPR store-data, data streamed straight from LDS.
        __builtin_amdgcn_global_store_async_from_lds_b128(
            (gv4i_ptr)(out4 + i), lsrc, /*offset=*/0, /*cpol=*/0);
    }
#if __has_builtin(__builtin_amdgcn_s_wait_asynccnt)
    __builtin_amdgcn_s_wait_asynccnt(0);
#endif
#else
    for (; i < n_vec4; i += stride) {
        out4[i] = z;
    }
#endif

    // Generic tail (out_size % 4). Zero for this problem shape.
    if (rem && blockIdx.x == 0 && tid < rem) {
        out[n_vec4 * 4 + tid] = 0.0f;
    }
}

__global__ void onehot_scatter_ones(const long long* __restrict__ labels,
                                    float* __restrict__ out,
                                    int rows, int C) {
    int r = blockIdx.x * blockDim.x + threadIdx.x;
    if (r >= rows) return;
    long long lab = labels[r];            // int64 labels in [-1, C)
    if (lab >= 0) {
        out[(long long)r * C + lab] = 1.0f;
    }
}

extern "C" void kernel_launch(void* const* d_in, const int* in_sizes, int n_in,
                              void* d_out, int out_size, void* d_ws, size_t ws_size,
                              hipStream_t stream) {
    // inputs: [0] obj_sem_cls_pred f32 [B,N,C] (unused by the math),
    //         [1] obj_labels int64 [B,N], [2] cur_step, [3] total_steps (unused)
    const long long* labels = (const long long*)d_in[1];
    float* out = (float*)d_out;

    const int rows = in_sizes[1];           // B*N = 32768
    const int C    = out_size / rows;       // 607

    const long long n_vec4 = (long long)out_size / 4;
    const int rem = out_size & 3;

    // Bulk zero fill: 2048 blocks x 256 threads, grid-stride over ~5M b128 chunks.
    dim3 fill_grid(2048), fill_block(256);
    onehot_zero_fill<<<fill_grid, fill_block, 0, stream>>>(out, n_vec4, rem);

    // Scatter the ones: one thread per (b,n) row. Stream order guarantees the
    // fill kernel (and its async stores) completed first.
    dim3 sc_grid((rows + 255) / 256), sc_block(256);
    onehot_scatter_ones<<<sc_grid, sc_block, 0, stream>>>(labels, out, rows, C);
}